// GaussianModel_18159121728141
// MI455X (gfx1250) — compile-verified
//
#include <hip/hip_runtime.h>
#include <hip/hip_bf16.h>

typedef __attribute__((ext_vector_type(2))) float v2f;
typedef __attribute__((ext_vector_type(8))) float v8f;

#define SEDGE 256
#define SCALEF 255.0f
#define TILE 16

// ---------------------------------------------------------------------------
// Zero-fill the 256^3 volume (d_out is poisoned; we accumulate, so zero it
// at the start of every launch). 64 MiB, 16B per thread.
// ---------------------------------------------------------------------------
__global__ void gs_zero_kernel(float4* __restrict__ p, int n4) {
    int i = blockIdx.x * blockDim.x + threadIdx.x;
    if (i < n4) {
        float4 z; z.x = 0.f; z.y = 0.f; z.z = 0.f; z.w = 0.f;
        p[i] = z;
    }
}

// ---------------------------------------------------------------------------
// One wave32 per Gaussian. Separable weights:
//   pass 1: lanes 0-15 -> ex[t], lanes 16-31 -> ey[t]   (t = lane & 15)
//   pass 2: all lanes   -> ez[t] (duplicated in both halves)
// Outer product ex (x) ey via V_WMMA_F32_16X16X4_F32 with only K=0 populated:
//   D(M,N) = ex[M] * ey[N], f32 exact.
// D layout: acc[v] @ lane l  ->  x = v + 8*(l>=16), y = l & 15.
// z-loop runs only over the valid z extent (wave-uniform trip count); per-v
// zero contributions (x/y bbox mask) skip their atomic.
// ---------------------------------------------------------------------------
__global__ void __launch_bounds__(256)
gs_splat_kernel(const float* __restrict__ centers,
                const float* __restrict__ sigmas,
                const float* __restrict__ intens,
                float* __restrict__ vol, int N) {
    const int lane = threadIdx.x & 31;
    const int g = blockIdx.x * 8 + (threadIdx.x >> 5);   // 8 waves / 256-thread block
    if (g >= N) return;                                  // wave-uniform: EXEC stays all-1s

    const float cx = centers[3 * g + 0];
    const float cy = centers[3 * g + 1];
    const float cz = centers[3 * g + 2];
    const float sg = sigmas[g];
    const float inten = intens[g];
    const float inv2s2 = 0.5f / (sg * sg);

    // Per-axis bbox exactly as the reference computes it.
    const float cvx = cx * SCALEF, cvy = cy * SCALEF, cvz = cz * SCALEF;
    const float cut = 3.0f * sg * SCALEF;
    const int mnx = (int)floorf(fmaxf(cvx - cut, 0.0f));
    const int mny = (int)floorf(fmaxf(cvy - cut, 0.0f));
    const int mnz = (int)floorf(fmaxf(cvz - cut, 0.0f));
    int mxx = (int)(floorf(fminf(cvx + cut, SCALEF)) + 1.0f); mxx = mxx > SEDGE ? SEDGE : mxx;
    int mxy = (int)(floorf(fminf(cvy + cut, SCALEF)) + 1.0f); mxy = mxy > SEDGE ? SEDGE : mxy;
    int mxz = (int)(floorf(fminf(cvz + cut, SCALEF)) + 1.0f); mxz = mxz > SEDGE ? SEDGE : mxz;

    const int t = lane & 15;
    const bool loHalf = lane < 16;

    // Pass 1: ex in lanes 0-15, ey in lanes 16-31 (invalid -> exact 0).
    const float ca  = loHalf ? cx : cy;
    const int   mna = loHalf ? mnx : mny;
    const int   mxa = loHalf ? mxx : mxy;
    float w01;
    {
        const int idx = mna + t;
        const float d = (float)idx * (1.0f / SCALEF) - ca;
        const float e = __expf(-d * d * inv2s2);
        w01 = (idx < mxa) ? e : 0.0f;
    }

    // Pass 2: ez[t] in every lane; exp never underflows inside the 3-sigma
    // cutoff, so validity is purely z < zcount (handled by the loop bound).
    float wz;
    {
        const int idx = mnz + t;
        const float d = (float)idx * (1.0f / SCALEF) - cz;
        wz = __expf(-d * d * inv2s2);
    }

    // Build WMMA operands: A col K=0 = ex, B row K=0 = ey, everything else 0.
    v2f A; A.x = loHalf ? w01 : 0.0f; A.y = 0.0f;
    const float eyv = __shfl(w01, t + 16, 32);           // ey[t] -> all lanes
    v2f B; B.x = loHalf ? eyv : 0.0f; B.y = 0.0f;
    v8f acc = {};
    acc = __builtin_amdgcn_wmma_f32_16x16x4_f32(
        /*neg_a=*/false, A, /*neg_b=*/false, B,
        /*c_mod=*/(short)0, acc, /*reuse_a=*/false, /*reuse_b=*/false);

    // Scatter: acc[v] = ex[xoff] * ey[y] with xoff = v + 8*(lane>=16), y = t.
    const int xbase = mnx + (loHalf ? 0 : 8);
    const int ybase = mny + t;
    // lin = x*65536 + y*256 + z
    const int laneBase = ybase * SEDGE + mnz;

    int zcount = mxz - mnz;                              // wave-uniform, <= 14
    zcount = zcount > TILE ? TILE : zcount;

#pragma unroll 1
    for (int z = 0; z < zcount; ++z) {
        // wave-uniform broadcast ez[z] via SGPR (v_readlane), no LDS op
        const float ezz =
            __uint_as_float(__builtin_amdgcn_readlane(__float_as_uint(wz), z));
        const float s = inten * ezz;
        const int zlin = laneBase + z;
#pragma unroll
        for (int v = 0; v < 8; ++v) {
            const float c = acc[v] * s;
            if (c != 0.0f) {                             // zero == masked by x/y bbox
                const int x = xbase + v;
                atomicAdd(&vol[x * (SEDGE * SEDGE) + zlin], c);
            }
        }
    }
}

extern "C" void kernel_launch(void* const* d_in, const int* in_sizes, int n_in,
                              void* d_out, int out_size, void* d_ws, size_t ws_size,
                              hipStream_t stream) {
    const float* centers = (const float*)d_in[0];
    const float* sigmas  = (const float*)d_in[1];
    const float* intens  = (const float*)d_in[2];
    float* vol = (float*)d_out;
    const int N = in_sizes[1];                           // sigmas: (N,)

    // 1) zero the volume (poisoned buffer + we accumulate)
    const int n4 = (SEDGE * SEDGE * SEDGE) / 4;
    gs_zero_kernel<<<(n4 + 255) / 256, 256, 0, stream>>>((float4*)vol, n4);

    // 2) splat: one wave32 per Gaussian, 8 waves per block
    const int blocks = (N + 7) / 8;
    gs_splat_kernel<<<blocks, 256, 0, stream>>>(centers, sigmas, intens, vol, N);
}